// SparseMiddleExtractor_73100343378285
// MI455X (gfx1250) — compile-verified
//
#include <hip/hip_runtime.h>
#include <hip/hip_bf16.h>

// ---------------- problem constants (match reference) ----------------
#define BB 2
#define D0 43
#define HH 200
#define WW 176
#define D1 21
#define D2 10
#define CIN 128
#define CC 64
#define EPSV 1e-3f

typedef __attribute__((ext_vector_type(2))) float v2f;
typedef __attribute__((ext_vector_type(8))) float v8f;

// ---------------- utility kernels ----------------
__global__ void fill_i32(int* __restrict__ p, int n, int val) {
    int stride = gridDim.x * blockDim.x;
    for (int i = blockIdx.x * blockDim.x + threadIdx.x; i < n; i += stride) p[i] = val;
}

__global__ void zero_f32(float* __restrict__ p, int n) {
    int stride = gridDim.x * blockDim.x;
    float4 z = {0.f, 0.f, 0.f, 0.f};
    for (int i = blockIdx.x * blockDim.x + threadIdx.x; i * 4 < n; i += stride)
        ((float4*)p)[i] = z;
}

__global__ void scatter_vol(const int* __restrict__ coords, int n,
                            int* __restrict__ vol, int D) {
    int i = blockIdx.x * blockDim.x + threadIdx.x;
    if (i >= n) return;
    int b = coords[i * 4 + 0];
    int d = coords[i * 4 + 1];
    int h = coords[i * 4 + 2];
    int w = coords[i * 4 + 3];
    vol[(((size_t)b * D + d) * HH + h) * WW + w] = i;
}

// 27-neighbor rulebook for submanifold 3x3x3 conv (dz outer, dy, dx inner)
__global__ void build_nbr27(const int* __restrict__ coords, int n,
                            const int* __restrict__ vol, int D,
                            int* __restrict__ nbr) {
    int gi = blockIdx.x * blockDim.x + threadIdx.x;
    if (gi >= n * 27) return;
    int i = gi / 27, k = gi % 27;
    int dz = k / 9 - 1;
    int dy = (k / 3) % 3 - 1;
    int dx = k % 3 - 1;
    int b = coords[i * 4 + 0];
    int d = coords[i * 4 + 1] + dz;
    int h = coords[i * 4 + 2] + dy;
    int w = coords[i * 4 + 3] + dx;
    int idx = -1;
    if (d >= 0 && d < D && h >= 0 && h < HH && w >= 0 && w < WW)
        idx = vol[(((size_t)b * D + d) * HH + h) * WW + w];
    nbr[gi] = idx;
}

// 3-tap rulebook for depth-strided conv: in(b, 2o+k, h, w)
__global__ void build_nbr3(const int* __restrict__ coords_out, int n,
                           const int* __restrict__ vol_in, int Din,
                           int* __restrict__ nbr) {
    int gi = blockIdx.x * blockDim.x + threadIdx.x;
    if (gi >= n * 3) return;
    int i = gi / 3, k = gi % 3;
    int b = coords_out[i * 4 + 0];
    int d = 2 * coords_out[i * 4 + 1] + k;
    int h = coords_out[i * 4 + 2];
    int w = coords_out[i * 4 + 3];
    int idx = -1;
    if (d >= 0 && d < Din)
        idx = vol_in[(((size_t)b * Din + d) * HH + h) * WW + w];
    nbr[gi] = idx;
}

// ---------------- gather-GEMM with fp32 WMMA ----------------
// out[nrows][64] = sum_kv gather(feats, nbr[:,kv]) @ weight[kv]   (weight: [KV][K][64])
// Block: 128 threads = 4 wave32. Tile: 32 rows x 64 cols.
// Each wave owns cols [16w,16w+16) and TWO 16x16 row tiles (rows 0-15, 16-31),
// sharing one pair of B registers per K-step across both WMMAs (halves B loads,
// and the two WMMAs are independent -> covers WMMA->WMMA RAW latency).
__global__ void conv_wmma(const float* __restrict__ feats,
                          const float* __restrict__ weight,
                          const int* __restrict__ nbr,
                          float* __restrict__ out,
                          int nrows, int KV, int K) {
    __shared__ float As[32 * 132];            // supports K up to 128 (ldsk = K+4)
    const int ldsk = K + 4;
    const int t = threadIdx.x;
    const int lane = t & 31;
    const int wv = t >> 5;                    // wave id 0..3
    const int col0 = wv * 16;
    const int n16 = lane & 15;                // N (and A-row) index within tile
    const int kh = (lane >> 4) * 2;           // K sub-offset: lanes 0-15 -> {0,1}, 16-31 -> {2,3}
    const int row0 = blockIdx.x * 32;

    v8f acc0 = {};
    v8f acc1 = {};

    for (int kv = 0; kv < KV; ++kv) {
        __syncthreads();
        // ---- stage A: 32 gathered feature rows into LDS (zeros for missing sites)
        {
            int r = t >> 2;                    // 0..31 (4 threads per row)
            int c0 = (t & 3) * 4;              // starting column, step 16
            int row = row0 + r;
            int idx = (row < nrows) ? nbr[(size_t)row * KV + kv] : -1;
            if (idx >= 0) {
                const float4* src = (const float4*)(feats + (size_t)idx * K);
                for (int c = c0; c < K; c += 16)
                    *(float4*)&As[r * ldsk + c] = src[c >> 2];
            } else {
                float4 z = {0.f, 0.f, 0.f, 0.f};
                for (int c = c0; c < K; c += 16)
                    *(float4*)&As[r * ldsk + c] = z;
            }
        }
        __syncthreads();

        // ---- K loop: V_WMMA_F32_16X16X4_F32, D = A(16x4) x B(4x16) + C
        const float* wk = weight + (size_t)kv * K * CC;
        for (int kk = 0; kk < K; kk += 4) {
            int kc = kk + kh;
            // B layout (4x16): lane n holds rows {kc, kc+1}, col n (shared by both row tiles)
            float b0 = wk[(size_t)(kc + 0) * CC + col0 + n16];
            float b1 = wk[(size_t)(kc + 1) * CC + col0 + n16];
            v2f b = {b0, b1};
            // A layout (32-bit 16x4): lane L(0-15): v0=K0,v1=K1 of row M=L; lane L+16: K2,K3
            float2 av0 = *(const float2*)&As[n16 * ldsk + kc];
            float2 av1 = *(const float2*)&As[(16 + n16) * ldsk + kc];
            v2f a0 = {av0.x, av0.y};
            v2f a1 = {av1.x, av1.y};
            acc0 = __builtin_amdgcn_wmma_f32_16x16x4_f32(
                false, a0, false, b, (short)0, acc0, false, false);
            acc1 = __builtin_amdgcn_wmma_f32_16x16x4_f32(
                false, a1, false, b, (short)0, acc1, false, false);
        }
    }

    // ---- store C/D: lane n (0-15): rows 0..7 col n in acc[0..7]; lane n+16: rows 8..15
    int mbase = (lane >> 4) * 8;
#pragma unroll
    for (int i = 0; i < 8; ++i) {
        int row = row0 + mbase + i;
        if (row < nrows) out[(size_t)row * CC + col0 + n16] = acc0[i];
    }
#pragma unroll
    for (int i = 0; i < 8; ++i) {
        int row = row0 + 16 + mbase + i;
        if (row < nrows) out[(size_t)row * CC + col0 + n16] = acc1[i];
    }
}

// ---------------- batchnorm (deterministic tree reduction) ----------------
// 64 blocks, one per channel; writes scale = g*rsqrt(var+eps), shift = b - mu*scale
__global__ void bn_stats(const float* __restrict__ x, int n,
                         const float* __restrict__ g, const float* __restrict__ b,
                         float* __restrict__ scale, float* __restrict__ shift) {
    __shared__ float ss[256];
    __shared__ float ss2[256];
    int c = blockIdx.x;
    float s = 0.f, s2 = 0.f;
    for (int r = threadIdx.x; r < n; r += 256) {
        float v = x[(size_t)r * CC + c];
        s += v;
        s2 += v * v;
    }
    ss[threadIdx.x] = s;
    ss2[threadIdx.x] = s2;
    __syncthreads();
    for (int off = 128; off > 0; off >>= 1) {
        if (threadIdx.x < off) {
            ss[threadIdx.x] += ss[threadIdx.x + off];
            ss2[threadIdx.x] += ss2[threadIdx.x + off];
        }
        __syncthreads();
    }
    if (threadIdx.x == 0) {
        float inv = 1.f / (float)n;
        float mu = ss[0] * inv;
        float var = ss2[0] * inv - mu * mu;
        float sc = g[c] * rsqrtf(var + EPSV);
        scale[c] = sc;
        shift[c] = b[c] - mu * sc;
    }
}

__global__ void bn_apply_relu(float* __restrict__ x, int total,
                              const float* __restrict__ scale,
                              const float* __restrict__ shift) {
    int i = blockIdx.x * blockDim.x + threadIdx.x;
    if (i >= total) return;
    int c = i & (CC - 1);
    float v = x[i] * scale[c] + shift[c];
    x[i] = v > 0.f ? v : 0.f;
}

// ---------------- final dense scatter: out[b][c*D2+d][h][w] ----------------
__global__ void scatter_out(const int* __restrict__ coords2, int n2,
                            const float* __restrict__ x, float* __restrict__ out) {
    int gi = blockIdx.x * blockDim.x + threadIdx.x;
    if (gi >= n2 * CC) return;
    int i = gi >> 6, c = gi & (CC - 1);
    int b = coords2[i * 4 + 0];
    int d = coords2[i * 4 + 1];
    int h = coords2[i * 4 + 2];
    int w = coords2[i * 4 + 3];
    out[((((size_t)b * CC + c) * D2 + d) * HH + h) * WW + w] = x[gi];
}

// ---------------- host orchestration ----------------
extern "C" void kernel_launch(void* const* d_in, const int* in_sizes, int n_in,
                              void* d_out, int out_size, void* d_ws, size_t ws_size,
                              hipStream_t stream) {
    const float* feat    = (const float*)d_in[0];
    const int*   coors   = (const int*)d_in[1];
    const int*   coords1 = (const int*)d_in[2];
    const int*   coords2 = (const int*)d_in[3];
    const float* w1_sub  = (const float*)d_in[5];
    const float* w1_down = (const float*)d_in[6];
    const float* w2a     = (const float*)d_in[7];
    const float* w2b     = (const float*)d_in[8];
    const float* w2_down = (const float*)d_in[9];
    const float* g1a = (const float*)d_in[10]; const float* b1a = (const float*)d_in[11];
    const float* g1b = (const float*)d_in[12]; const float* b1b = (const float*)d_in[13];
    const float* g2a = (const float*)d_in[14]; const float* b2a = (const float*)d_in[15];
    const float* g2b = (const float*)d_in[16]; const float* b2b = (const float*)d_in[17];
    const float* g2c = (const float*)d_in[18]; const float* b2c = (const float*)d_in[19];

    const int N0 = in_sizes[1] / 4;
    const int N1 = in_sizes[2] / 4;
    const int N2 = in_sizes[3] / 4;
    int nmax = N0 > N1 ? N0 : N1;
    if (N2 > nmax) nmax = N2;

    const int VOL0 = BB * D0 * HH * WW;
    const int VOL1 = BB * D1 * HH * WW;
    int nbr_max = N0 * 27;
    if (N1 * 27 > nbr_max) nbr_max = N1 * 27;
    if (N2 * 3  > nbr_max) nbr_max = N2 * 3;

    // workspace carve (256B aligned slabs)
    size_t off = 0;
    char* base = (char*)d_ws;
    auto carve = [&](size_t bytes) -> char* {
        char* p = base + off;
        off += (bytes + 255) & ~(size_t)255;
        return p;
    };
    int*   vol0  = (int*)  carve((size_t)VOL0 * 4);
    int*   vol1  = (int*)  carve((size_t)VOL1 * 4);
    int*   nbr   = (int*)  carve((size_t)nbr_max * 4);
    float* xa    = (float*)carve((size_t)nmax * CC * 4);
    float* xb    = (float*)carve((size_t)nmax * CC * 4);
    float* scale = (float*)carve(CC * 4);
    float* shift = (float*)carve(CC * 4);
    (void)ws_size; (void)n_in;

    const int T = 256;
    auto cdiv = [](int a, int b) { return (a + b - 1) / b; };

    // ---- stage 0: build vol0
    fill_i32<<<2048, T, 0, stream>>>(vol0, VOL0, -1);
    scatter_vol<<<cdiv(N0, T), T, 0, stream>>>(coors, N0, vol0, D0);

    // ---- conv1 (subm 3x3x3, 128 -> 64) + BN + ReLU
    build_nbr27<<<cdiv(N0 * 27, T), T, 0, stream>>>(coors, N0, vol0, D0, nbr);
    conv_wmma<<<cdiv(N0, 32), 128, 0, stream>>>(feat, w1_sub, nbr, xa, N0, 27, CIN);
    bn_stats<<<CC, T, 0, stream>>>(xa, N0, g1a, b1a, scale, shift);
    bn_apply_relu<<<cdiv(N0 * CC, T), T, 0, stream>>>(xa, N0 * CC, scale, shift);

    // ---- down1 (depth conv 3, stride 2) + BN + ReLU
    build_nbr3<<<cdiv(N1 * 3, T), T, 0, stream>>>(coords1, N1, vol0, D0, nbr);
    conv_wmma<<<cdiv(N1, 32), 128, 0, stream>>>(xa, w1_down, nbr, xb, N1, 3, CC);
    bn_stats<<<CC, T, 0, stream>>>(xb, N1, g1b, b1b, scale, shift);
    bn_apply_relu<<<cdiv(N1 * CC, T), T, 0, stream>>>(xb, N1 * CC, scale, shift);

    // ---- stage 1 volume + shared subm rulebook (same for w2a and w2b)
    fill_i32<<<2048, T, 0, stream>>>(vol1, VOL1, -1);
    scatter_vol<<<cdiv(N1, T), T, 0, stream>>>(coords1, N1, vol1, D1);
    build_nbr27<<<cdiv(N1 * 27, T), T, 0, stream>>>(coords1, N1, vol1, D1, nbr);

    // ---- conv2a + BN + ReLU
    conv_wmma<<<cdiv(N1, 32), 128, 0, stream>>>(xb, w2a, nbr, xa, N1, 27, CC);
    bn_stats<<<CC, T, 0, stream>>>(xa, N1, g2a, b2a, scale, shift);
    bn_apply_relu<<<cdiv(N1 * CC, T), T, 0, stream>>>(xa, N1 * CC, scale, shift);

    // ---- conv2b + BN + ReLU (reuses nbr)
    conv_wmma<<<cdiv(N1, 32), 128, 0, stream>>>(xa, w2b, nbr, xb, N1, 27, CC);
    bn_stats<<<CC, T, 0, stream>>>(xb, N1, g2b, b2b, scale, shift);
    bn_apply_relu<<<cdiv(N1 * CC, T), T, 0, stream>>>(xb, N1 * CC, scale, shift);

    // ---- down2 + BN + ReLU
    build_nbr3<<<cdiv(N2 * 3, T), T, 0, stream>>>(coords2, N2, vol1, D1, nbr);
    conv_wmma<<<cdiv(N2, 32), 128, 0, stream>>>(xb, w2_down, nbr, xa, N2, 3, CC);
    bn_stats<<<CC, T, 0, stream>>>(xa, N2, g2c, b2c, scale, shift);
    bn_apply_relu<<<cdiv(N2 * CC, T), T, 0, stream>>>(xa, N2 * CC, scale, shift);

    // ---- densify
    zero_f32<<<4096, T, 0, stream>>>((float*)d_out, out_size);
    scatter_out<<<cdiv(N2 * CC, T), T, 0, stream>>>(coords2, N2, xa, (float*)d_out);
}